// MultiHeadSelfAttention_13993003450569
// MI455X (gfx1250) — compile-verified
//
#include <hip/hip_runtime.h>
#include <hip/hip_bf16.h>

// CDNA5 fp32 WMMA fragments: A/B of 16x16x4 f32 are 2 VGPRs, C/D are 8 VGPRs.
typedef __attribute__((ext_vector_type(2))) float v2f;
typedef __attribute__((ext_vector_type(8))) float v8f;

// Problem constants (reference: B=2, S=2048, D=1024, H=16, d=64)
#define BATCH 2
#define SEQ   2048
#define DIM   1024
#define NHEAD 16
#define HDIM  64
#define HEADMAT ((size_t)SEQ * HDIM)              // 131072 floats per (b,h) matrix
#define QKV_ONE ((size_t)BATCH * NHEAD * HEADMAT) // 4194304 floats per Q/K/V buffer

static __device__ __forceinline__ v8f wmma_f32(v2f a, v2f b, v8f c) {
    // D = A(16x4) * B(4x16) + C on the fp32 matrix pipe
    return __builtin_amdgcn_wmma_f32_16x16x4_f32(false, a, false, b, (short)0, c,
                                                 false, false);
}

// ---------------------------------------------------------------------------
// Kernel 1: kqv = X @ W + b, scattered into workspace as K,Q,V in [B,H,S,d].
// X: [4096,1024], W: [1024,3072] (cols = c*1024 + h*64 + dd, c in {K,Q,V}).
// Block tile 128(M) x 128(N), 8 waves; wave tile 32(M) x 64(N): every B-frag
// feeds two WMMAs (2 M-subtiles), halving LDS operand traffic per FLOP.
// ---------------------------------------------------------------------------
__global__ __launch_bounds__(256) void kqv_gemm_kernel(
    const float* __restrict__ X, const float* __restrict__ W,
    const float* __restrict__ bias, float* __restrict__ qkv) {
    __shared__ float As[128][36];   // 128x32 K-chunk of X (36%4==0: b128 aligned)
    __shared__ float Bs[32][132];   // 32x128 K-chunk of W

    const int tid  = threadIdx.x;
    const int wave = tid >> 5;
    const int lane = tid & 31;
    const int n    = lane & 15;     // tile column / A-row
    const int half = lane >> 4;     // K-half selector (ISA f32 A/B layout)

    const int mBlock = (blockIdx.x % 32) * 128;  // grid.x = 32 * 24
    const int nBlock = (blockIdx.x / 32) * 128;
    const int mOff   = (wave & 3) * 32;          // 4 M-slots of 32 rows
    const int nOff   = (wave >> 2) * 64;         // 2 N-slots of 64 cols

    v8f acc[2][4] = {};

    for (int kc = 0; kc < DIM; kc += 32) {
        __syncthreads();
        // Stage X tile: 128x32 = 1024 float4, 4 per thread (coalesced rows)
        #pragma unroll
        for (int i = 0; i < 4; ++i) {
            int idx = tid + i * 256;
            int r = idx >> 3, c4 = (idx & 7) * 4;
            *(float4*)&As[r][c4] =
                *(const float4*)(X + (size_t)(mBlock + r) * DIM + kc + c4);
            if (kc + 32 < DIM)  // hide next-chunk latency (global_prefetch_b8)
                __builtin_prefetch(X + (size_t)(mBlock + r) * DIM + kc + 32 + c4, 0, 0);
        }
        // Stage W tile: 32x128 = 1024 float4, 4 per thread
        #pragma unroll
        for (int i = 0; i < 4; ++i) {
            int idx = tid + i * 256;
            int r = idx >> 5, c4 = (idx & 31) * 4;
            *(float4*)&Bs[r][c4] =
                *(const float4*)(W + (size_t)(kc + r) * (3 * DIM) + nBlock + c4);
            if (kc + 32 < DIM)
                __builtin_prefetch(W + (size_t)(kc + 32 + r) * (3 * DIM) + nBlock + c4, 0, 0);
        }
        __syncthreads();

        #pragma unroll
        for (int kk = 0; kk < 32; kk += 4) {
            // A-frags for both M-subtiles: contiguous float2 per lane
            v2f a0 = *(const v2f*)&As[mOff + n][kk + 2 * half];
            v2f a1 = *(const v2f*)&As[mOff + 16 + n][kk + 2 * half];
            #pragma unroll
            for (int ct = 0; ct < 4; ++ct) {
                v2f b;
                b.x = Bs[kk + 2 * half][nOff + ct * 16 + n];
                b.y = Bs[kk + 2 * half + 1][nOff + ct * 16 + n];
                acc[0][ct] = wmma_f32(a0, b, acc[0][ct]);  // B-frag reused twice
                acc[1][ct] = wmma_f32(a1, b, acc[1][ct]);
            }
        }
    }

    // Epilogue: add bias, scatter to [B,H,S,d] K/Q/V buffers.
    #pragma unroll
    for (int ct = 0; ct < 4; ++ct) {
        const int ng  = nBlock + nOff + ct * 16 + n;   // global column in [0,3072)
        const int c   = ng >> 10;                      // 0=K, 1=Q, 2=V
        const int rem = ng & 1023;
        const int h   = rem >> 6;
        const int dd  = rem & 63;
        const float bv = bias[ng];
        float* base = qkv + (size_t)c * QKV_ONE;
        #pragma unroll
        for (int r = 0; r < 2; ++r) {
            #pragma unroll
            for (int j = 0; j < 8; ++j) {
                // C-layout: VGPR j -> row j (lanes 0-15) / j+8 (lanes 16-31)
                const int mg = mBlock + mOff + r * 16 + j + 8 * half;
                const int bb = mg >> 11;          // batch
                const int s  = mg & 2047;         // sequence position
                base[((size_t)(bb * NHEAD + h) * SEQ + s) * HDIM + dd] =
                    acc[r][ct][j] + bv;
            }
        }
    }
}

// ---------------------------------------------------------------------------
// Kernel 2: fused flash-style attention per (b, h, q-tile of 128 rows).
//   u = Q Kᵀ ; w = exp(min(elu(u),10)) * mask_k ; num += w V ; den += w 1
//   out = mask_q * num / den
// 128 threads = 4 waves x 32 query rows: every K/V B-fragment feeds two
// WMMAs. Denominator via an extra WMMA with a constant all-ones B-fragment,
// so num and den share the C-layout and the final division is lane-local.
// ---------------------------------------------------------------------------
__global__ __launch_bounds__(128) void attention_kernel(
    const float* __restrict__ qkv, const float* __restrict__ mask,
    float* __restrict__ out) {
    __shared__ float Ks[64][68];         // 64-key chunk of K (row=key, col=d)
    __shared__ float Vs[64][68];         // 64-key chunk of V
    __shared__ float Ws[4][2][16][18];   // per-wave 2x(16x16) w tiles (C->A)

    const int tid  = threadIdx.x;
    const int wave = tid >> 5;           // 0..3
    const int lane = tid & 31;
    const int n    = lane & 15;
    const int half = lane >> 4;

    const int bid   = blockIdx.x;        // 512 = B * H * (S/128)
    const int qTile = bid & 15;
    const int h     = (bid >> 4) & 15;
    const int b     = bid >> 8;

    const float* Kp = qkv;
    const float* Qp = qkv + QKV_ONE;
    const float* Vp = qkv + 2 * QKV_ONE;
    const size_t headOff = (size_t)(b * NHEAD + h) * HEADMAT;
    const int qBase = qTile * 128 + wave * 32;   // 32 q rows per wave

    // Pre-build Q A-fragments for both 16-row subtiles: 2 x 16 K-steps (d=64).
    v2f qf[2][16];
    #pragma unroll
    for (int r = 0; r < 2; ++r)
        #pragma unroll
        for (int t = 0; t < 16; ++t)
            qf[r][t] = *(const v2f*)(Qp + headOff +
                                     (size_t)(qBase + r * 16 + n) * HDIM +
                                     t * 4 + 2 * half);

    v8f num[2][4] = {};
    v8f den[2]    = {};
    const v2f ones = {1.0f, 1.0f};

    for (int kb = 0; kb < SEQ; kb += 64) {
        __syncthreads();
        // Cooperative K/V chunk staging: 64x64 each = 1024 float4 apiece.
        #pragma unroll
        for (int i = 0; i < 8; ++i) {
            int idx = tid + i * 128;
            int r = idx >> 4, c4 = (idx & 15) * 4;
            *(float4*)&Ks[r][c4] =
                *(const float4*)(Kp + headOff + (size_t)(kb + r) * HDIM + c4);
            *(float4*)&Vs[r][c4] =
                *(const float4*)(Vp + headOff + (size_t)(kb + r) * HDIM + c4);
            if (kb + 64 < SEQ) {   // prefetch next chunk (global_prefetch_b8)
                __builtin_prefetch(Kp + headOff + (size_t)(kb + 64 + r) * HDIM + c4, 0, 0);
                __builtin_prefetch(Vp + headOff + (size_t)(kb + 64 + r) * HDIM + c4, 0, 0);
            }
        }
        __syncthreads();

        #pragma unroll
        for (int kt = 0; kt < 4; ++kt) {
            // ---- u = Q Kᵀ for this 16-key subtile, both row-blocks ----
            v8f u0 = {}, u1 = {};
            #pragma unroll
            for (int t = 0; t < 16; ++t) {
                // B[d][key]: lane n supplies key (kt*16+n), d pair (4t+2*half)
                v2f bk = *(const v2f*)&Ks[kt * 16 + n][t * 4 + 2 * half];
                u0 = wmma_f32(qf[0][t], bk, u0);   // bk reused twice
                u1 = wmma_f32(qf[1][t], bk, u1);
            }
            // ---- elementwise: w = exp(min(elu(u),10)) * mask_k ----
            const float mk = mask[b * SEQ + kb + kt * 16 + n];
            #pragma unroll
            for (int j = 0; j < 8; ++j) {
                float uu = u0[j];
                float e  = uu > 0.0f ? fminf(uu, 10.0f) : (expf(uu) - 1.0f);
                Ws[wave][0][j + 8 * half][n] = expf(e) * mk;  // [q-row][key]
                uu = u1[j];
                e  = uu > 0.0f ? fminf(uu, 10.0f) : (expf(uu) - 1.0f);
                Ws[wave][1][j + 8 * half][n] = expf(e) * mk;
            }
            // Same-wave DS ops are in order; A-frag reads below see the stores.
            #pragma unroll
            for (int kk = 0; kk < 4; ++kk) {
                v2f a0 = *(const v2f*)&Ws[wave][0][n][kk * 4 + 2 * half];
                v2f a1 = *(const v2f*)&Ws[wave][1][n][kk * 4 + 2 * half];
                #pragma unroll
                for (int nt = 0; nt < 4; ++nt) {
                    v2f bv;
                    bv.x = Vs[kt * 16 + kk * 4 + 2 * half][nt * 16 + n];
                    bv.y = Vs[kt * 16 + kk * 4 + 2 * half + 1][nt * 16 + n];
                    num[0][nt] = wmma_f32(a0, bv, num[0][nt]);  // bv reused twice
                    num[1][nt] = wmma_f32(a1, bv, num[1][nt]);
                }
                den[0] = wmma_f32(a0, ones, den[0]);  // row sums, lane-local
                den[1] = wmma_f32(a1, ones, den[1]);
            }
        }
    }

    // ---- epilogue: out[b, q, h*64 + d] = mask_q * num / den ----
    #pragma unroll
    for (int r = 0; r < 2; ++r) {
        #pragma unroll
        for (int j = 0; j < 8; ++j) {
            const int q   = qBase + r * 16 + j + 8 * half;
            const float f = mask[b * SEQ + q] / den[r][j];
            #pragma unroll
            for (int nt = 0; nt < 4; ++nt)
                out[(size_t)(b * SEQ + q) * DIM + h * HDIM + nt * 16 + n] =
                    num[r][nt][j] * f;
        }
    }
}

extern "C" void kernel_launch(void* const* d_in, const int* in_sizes, int n_in,
                              void* d_out, int out_size, void* d_ws, size_t ws_size,
                              hipStream_t stream) {
    (void)in_sizes; (void)n_in; (void)out_size; (void)ws_size;
    const float* X    = (const float*)d_in[0];   // inputs [2,2048,1024]
    const float* mask = (const float*)d_in[1];   // masks  [2,2048]
    const float* W    = (const float*)d_in[2];   // W      [1024,3072]
    const float* bias = (const float*)d_in[3];   // b      [3072]
    float* out = (float*)d_out;                  // [2,2048,1024]
    float* qkv = (float*)d_ws;                   // 3 * 4194304 floats (48 MB)

    // GEMM1: 4096x3072x1024, block tile 128x128 -> 32*24 blocks
    kqv_gemm_kernel<<<dim3(32 * 24), 256, 0, stream>>>(X, W, bias, qkv);
    // Fused attention: B*H*(S/128) = 512 blocks of 128 threads
    attention_kernel<<<dim3(512), 128, 0, stream>>>(qkv, mask, out);
}